// nBRCModel_77481210020230
// MI455X (gfx1250) — compile-verified
//
#include <hip/hip_runtime.h>
#include <math.h>

// ---------------------------------------------------------------------------
// nBRC model on gfx1250 (MI455X), full f32 precision via V_WMMA_F32_16X16X4_F32.
//
// Input flattening assumption (depth-first, dict insertion order):
//  [0] x [64,512,256]
//  layer l base = 1 + 10*l : Ua_w, Ua_b, Wa_w, Wa_b, Uc_w, Uc_b, Wc_w, Wc_b, U_w, U_b
//  [21] fc_w [512,5], [22] fc_b [5]
// ---------------------------------------------------------------------------

typedef __attribute__((ext_vector_type(2))) float v2f;
typedef __attribute__((ext_vector_type(4))) float v4f;
typedef __attribute__((ext_vector_type(8))) float v8f;

// Only check for the builtin on the device pass; on the host pass clang
// targets x86 and __has_builtin(...) for amdgcn builtins is always false.
#if defined(__HIP_DEVICE_COMPILE__) && !__has_builtin(__builtin_amdgcn_wmma_f32_16x16x4_f32)
#error "missing __builtin_amdgcn_wmma_f32_16x16x4_f32 on this toolchain"
#endif

// D = A(16x4 f32) x B(4x16 f32) + C(16x16 f32)
__device__ inline v8f wmma_f32(v2f a, v2f b, v8f c) {
  return __builtin_amdgcn_wmma_f32_16x16x4_f32(
      /*neg_a=*/false, a, /*neg_b=*/false, b,
      /*c_mod=*/(short)0, c, /*reuse_a=*/false, /*reuse_b=*/false);
}

static constexpr int Bsz = 64;
static constexpr int T   = 512;
static constexpr int H   = 512;
static constexpr int D0  = 256;
static constexpr int M   = Bsz * T;   // 32768 tokens

// ---------------------------------------------------------------------------
// Fused triple-projection GEMM: oa/oc/ou = X @ {Wa,Wc,Wu} + {ba,bc,bu}
// X: [M, K] row-major, W*: [K, 512], o*: [M, 512].
// One wave per 16x16 output tile; A loads shared across the three gates.
// grid = (H/16, M/128), block = 256 (8 waves -> 8 M-tiles per block).
// ---------------------------------------------------------------------------
__global__ __launch_bounds__(256) void proj3_kernel(
    const float* __restrict__ X, int K,
    const float* __restrict__ Wa, const float* __restrict__ ba,
    const float* __restrict__ Wc, const float* __restrict__ bc,
    const float* __restrict__ Wu, const float* __restrict__ bu,
    float* __restrict__ oa, float* __restrict__ oc, float* __restrict__ ou) {
  const int lane  = threadIdx.x & 31;
  const int wave  = threadIdx.x >> 5;
  const int khalf = lane >> 4;      // 0: K{0,1}, 1: K{2,3} of each k-step
  const int l16   = lane & 15;
  const int m0 = (blockIdx.y * 8 + wave) * 16;
  const int n0 = blockIdx.x * 16;
  const int col = n0 + l16;

  const float* xrow = X + (size_t)(m0 + l16) * K;

  v8f ca = {}, cc = {}, cu = {};
#pragma unroll 4
  for (int k = 0; k < K; k += 4) {
    const int r0 = k + 2 * khalf;
    v2f A = *(const v2f*)(xrow + r0);          // global_load_b64, b128-coalesced
    v2f Ba, Bc2, Bu2;
    Ba.x  = Wa[r0 * H + col];  Ba.y  = Wa[(r0 + 1) * H + col];
    Bc2.x = Wc[r0 * H + col];  Bc2.y = Wc[(r0 + 1) * H + col];
    Bu2.x = Wu[r0 * H + col];  Bu2.y = Wu[(r0 + 1) * H + col];
    ca = wmma_f32(A, Ba,  ca);
    cc = wmma_f32(A, Bc2, cc);
    cu = wmma_f32(A, Bu2, cu);
  }

  const float bav = ba[col], bcv = bc[col], buv = bu[col];
#pragma unroll
  for (int i = 0; i < 8; ++i) {
    // D layout: VGPR i -> M = i (lanes 0-15) / i+8 (lanes 16-31), N = lane%16
    const int row = m0 + i + 8 * khalf;
    const int idx = row * H + col;
    oa[idx] = ca[i] + bav;
    oc[idx] = cc[i] + bcv;
    ou[idx] = cu[i] + buv;
  }
}

// ---------------------------------------------------------------------------
// nBRC recurrent scan. One workgroup per 16-row batch block (grid = 4),
// 512 threads = 16 waves; wave w owns hidden columns [32w, 32w+32).
// h (16x512 f32 = 32KB) lives in LDS; Wa/Wc stream from L2 every step.
// Per step: WMMA K-loop over LDS h  ->  barrier  ->  elementwise update of
// exclusively-owned h elements + store y  ->  barrier.
// ---------------------------------------------------------------------------
__global__ __launch_bounds__(512) void nbrc_scan_kernel(
    const float* __restrict__ pa, const float* __restrict__ pc,
    const float* __restrict__ pu,
    const float* __restrict__ Wa, const float* __restrict__ ba,
    const float* __restrict__ Wc, const float* __restrict__ bc,
    float* __restrict__ y) {
  __shared__ float hbuf[16 * H];   // 32 KB

  const int tid   = threadIdx.x;
  const int lane  = tid & 31;
  const int wave  = tid >> 5;
  const int khalf = lane >> 4;
  const int l16   = lane & 15;
  const int b0    = blockIdx.x * 16;
  const int nb    = wave * 32;

  for (int i = tid; i < 16 * H; i += 512) hbuf[i] = 0.0f;
  __syncthreads();

  const int col0 = nb + l16;
  const int col1 = nb + 16 + l16;
  const float ba0 = ba[col0], ba1 = ba[col1];
  const float bc0 = bc[col0], bc1 = bc[col1];
  const float* hrow = hbuf + l16 * H;

  for (int t = 0; t < T; ++t) {
    v8f aa0 = {}, aa1 = {}, cc0 = {}, cc1 = {};
#pragma unroll 4
    for (int k = 0; k < H; k += 4) {
      const int r0 = k + 2 * khalf;
      v2f A = *(const v2f*)(hrow + r0);        // ds_load_b64 from LDS h
      v2f B0, B1, B2, B3;
      B0.x = Wa[r0 * H + col0];  B0.y = Wa[(r0 + 1) * H + col0];
      B1.x = Wa[r0 * H + col1];  B1.y = Wa[(r0 + 1) * H + col1];
      B2.x = Wc[r0 * H + col0];  B2.y = Wc[(r0 + 1) * H + col0];
      B3.x = Wc[r0 * H + col1];  B3.y = Wc[(r0 + 1) * H + col1];
      aa0 = wmma_f32(A, B0, aa0);
      aa1 = wmma_f32(A, B1, aa1);
      cc0 = wmma_f32(A, B2, cc0);
      cc1 = wmma_f32(A, B3, cc1);
    }
    __syncthreads();   // all waves finished reading h for this step

#pragma unroll
    for (int i = 0; i < 8; ++i) {
      const int rl = i + 8 * khalf;                       // local batch row
      const int g  = ((b0 + rl) * T + t) * H;
      {
        const float hold = hbuf[rl * H + col0];
        const float a = 1.0f + tanhf(aa0[i] + pa[g + col0] + ba0);
        const float s = 1.0f / (1.0f + __expf(-(cc0[i] + pc[g + col0] + bc0)));
        const float hn = s * hold + (1.0f - s) * tanhf(pu[g + col0] + a * hold);
        hbuf[rl * H + col0] = hn;                         // exclusively owned
        y[g + col0] = hn;
      }
      {
        const float hold = hbuf[rl * H + col1];
        const float a = 1.0f + tanhf(aa1[i] + pa[g + col1] + ba1);
        const float s = 1.0f / (1.0f + __expf(-(cc1[i] + pc[g + col1] + bc1)));
        const float hn = s * hold + (1.0f - s) * tanhf(pu[g + col1] + a * hold);
        hbuf[rl * H + col1] = hn;
        y[g + col1] = hn;
      }
    }
    __syncthreads();   // h updates visible before next step's WMMA reads
  }
}

// ---------------------------------------------------------------------------
// Final FC: out[m, 0..4] = y[m, :] @ fc_w + fc_b.  N=5 -> plain VALU dot,
// fc_w staged in LDS. grid = M/256, block = 256, one token per thread.
// ---------------------------------------------------------------------------
__global__ __launch_bounds__(256) void fc_kernel(
    const float* __restrict__ y, const float* __restrict__ W,
    const float* __restrict__ b, float* __restrict__ out) {
  __shared__ float ws[H * 5];
  for (int i = threadIdx.x; i < H * 5; i += 256) ws[i] = W[i];
  __syncthreads();

  const int m = blockIdx.x * 256 + threadIdx.x;
  const float* yr = y + (size_t)m * H;
  float s0 = b[0], s1 = b[1], s2 = b[2], s3 = b[3], s4 = b[4];
#pragma unroll 2
  for (int k = 0; k < H; k += 4) {
    v4f yv = *(const v4f*)(yr + k);            // global_load_b128
#pragma unroll
    for (int q = 0; q < 4; ++q) {
      const float xv = yv[q];
      const float* wr = ws + (k + q) * 5;
      s0 += xv * wr[0]; s1 += xv * wr[1]; s2 += xv * wr[2];
      s3 += xv * wr[3]; s4 += xv * wr[4];
    }
  }
  float* o = out + (size_t)m * 5;
  o[0] = s0; o[1] = s1; o[2] = s2; o[3] = s3; o[4] = s4;
}

// ---------------------------------------------------------------------------
extern "C" void kernel_launch(void* const* d_in, const int* in_sizes, int n_in,
                              void* d_out, int out_size, void* d_ws, size_t ws_size,
                              hipStream_t stream) {
  auto F = [&](int i) { return (const float*)d_in[i]; };
  const float* x = F(0);

  float* wsf = (float*)d_ws;
  const size_t G = (size_t)M * H;              // 16,777,216 floats per buffer
  float* ua = wsf;
  float* uc = wsf + G;
  float* uu = wsf + 2 * G;
  float* y  = wsf + 3 * G;                     // layer output / FC input

  const dim3 pg(H / 16, M / 128);              // (32, 256)

  // ---- Layer 1 (Din = 256) ----
  proj3_kernel<<<pg, 256, 0, stream>>>(x, D0,
      F(1), F(2),    // Ua_w, Ua_b
      F(5), F(6),    // Uc_w, Uc_b
      F(9), F(10),   // U_w,  U_b
      ua, uc, uu);
  nbrc_scan_kernel<<<4, 512, 0, stream>>>(ua, uc, uu,
      F(3), F(4),    // Wa_w, Wa_b
      F(7), F(8),    // Wc_w, Wc_b
      y);

  // ---- Layer 2 (Din = 512), input = y, output overwrites y ----
  proj3_kernel<<<pg, 256, 0, stream>>>(y, H,
      F(11), F(12),
      F(15), F(16),
      F(19), F(20),
      ua, uc, uu);
  nbrc_scan_kernel<<<4, 512, 0, stream>>>(ua, uc, uu,
      F(13), F(14),
      F(17), F(18),
      y);

  // ---- Final FC ----
  fc_kernel<<<M / 256, 256, 0, stream>>>(y, F(21), F(22), (float*)d_out);
}